// GCNConvRnd_67499706024646
// MI455X (gfx1250) — compile-verified
//
#include <hip/hip_runtime.h>
#include <hip/hip_bf16.h>

typedef __attribute__((ext_vector_type(2))) float v2f;
typedef __attribute__((ext_vector_type(8))) float v8f;

#define F_DIM 128

// ---------------------------------------------------------------------------
// Phase 1: h = x @ W.T  using V_WMMA_F32_16X16X4_F32 (fp32-exact, matrix pipe)
// One wave computes one 16x16 tile of h. blockDim = 256 (8 waves), each wave
// owns one of the 8 N-tiles; all waves in the block share the same 16 x-rows.
// ---------------------------------------------------------------------------
__global__ void __launch_bounds__(256)
gcn_gemm_wmma_f32(const float* __restrict__ x,   // [n_nodes, 128] row-major
                  const float* __restrict__ W,   // [128, 128] row-major (h = x*W^T)
                  float* __restrict__ h)         // [n_nodes, 128]
{
    const int lane  = threadIdx.x & 31;
    const int wave  = threadIdx.x >> 5;        // 0..7 -> N tile
    const int tileM = blockIdx.x;              // 0..(n_nodes/16 - 1)
    const int l15   = lane & 15;
    const int hi    = lane >> 4;               // 0: K pair {0,1}; 1: K pair {2,3}

    // A fragment: lane holds row m = tileM*16 + l15; v2f = x[m][k0+2*hi .. +1]
    const float* __restrict__ xrow = x + (size_t)(tileM * 16 + l15) * F_DIM;
    // B fragment: lane holds col n = wave*16 + l15; B[k][n] = W[n][k] contiguous
    const float* __restrict__ wrow = W + (size_t)(wave * 16 + l15) * F_DIM;

    v8f c = {};
#pragma unroll
    for (int k0 = 0; k0 < F_DIM; k0 += 4) {
        v2f a = *(const v2f*)(xrow + k0 + 2 * hi);
        v2f b = *(const v2f*)(wrow + k0 + 2 * hi);
        // 8 args: (neg_a, A, neg_b, B, c_mod, C, reuse_a, reuse_b)
        c = __builtin_amdgcn_wmma_f32_16x16x4_f32(false, a, false, b,
                                                  (short)0, c, false, false);
    }

    // C/D layout: VGPR r -> row M = 8*hi + r, col N = l15
    float* __restrict__ hp = h + (size_t)(tileM * 16 + hi * 8) * F_DIM + wave * 16 + l15;
#pragma unroll
    for (int r = 0; r < 8; ++r)
        hp[(size_t)r * F_DIM] = c[r];
}

// ---------------------------------------------------------------------------
// Phase 2: out[n, f] = bias[f]   (fold GCN bias into the accumulator init;
// d_out is poisoned by the harness, so this also serves as the zero-init)
// ---------------------------------------------------------------------------
__global__ void gcn_bias_init(const float* __restrict__ bias,
                              float* __restrict__ out, int total)
{
    int i = blockIdx.x * blockDim.x + threadIdx.x;
    if (i < total) out[i] = bias[i & (F_DIM - 1)];
}

// ---------------------------------------------------------------------------
// Phase 3: for each kept edge: out[dst] += h[src] * w
// One wave per edge; lane -> float4 slice (gather from L2-resident h, then
// 4x global_atomic_add_f32 resolving at the L2 atomic units).
// ---------------------------------------------------------------------------
__global__ void __launch_bounds__(256)
gcn_edge_scatter(const float* __restrict__ h,
                 const int*   __restrict__ edge_src,
                 const int*   __restrict__ edge_dst,
                 const float* __restrict__ edge_weight,
                 const int*   __restrict__ idx_keep,
                 float*       __restrict__ out,
                 int n_keep)
{
    const int e = blockIdx.x * (blockDim.x >> 5) + (threadIdx.x >> 5);
    if (e >= n_keep) return;
    const int lane = threadIdx.x & 31;

    const int   k   = idx_keep[e];
    const int   src = edge_src[k];
    const int   dst = edge_dst[k];
    const float w   = edge_weight[k];

    const float4 v = *(const float4*)(h + (size_t)src * F_DIM + lane * 4);
    float* __restrict__ o = out + (size_t)dst * F_DIM + lane * 4;
    unsafeAtomicAdd(o + 0, v.x * w);
    unsafeAtomicAdd(o + 1, v.y * w);
    unsafeAtomicAdd(o + 2, v.z * w);
    unsafeAtomicAdd(o + 3, v.w * w);
}

extern "C" void kernel_launch(void* const* d_in, const int* in_sizes, int n_in,
                              void* d_out, int out_size, void* d_ws, size_t ws_size,
                              hipStream_t stream)
{
    const float* x      = (const float*)d_in[0];   // [n_nodes,128]
    const float* W      = (const float*)d_in[1];   // [128,128]
    const float* bias   = (const float*)d_in[2];   // [128]
    const int*   esrc   = (const int*)  d_in[3];
    const int*   edst   = (const int*)  d_in[4];
    const float* ew     = (const float*)d_in[5];
    const int*   keep   = (const int*)  d_in[6];
    float*       out    = (float*)d_out;
    float*       h      = (float*)d_ws;            // [n_nodes,128] scratch (51.2 MB)

    const int n_nodes = in_sizes[0] / F_DIM;       // 100000 (divisible by 16)
    const int n_keep  = in_sizes[6];               // 1,280,000
    const int total   = n_nodes * F_DIM;

    // Phase 1: GEMM (one block per 16-row stripe, 8 waves cover all 8 N tiles)
    gcn_gemm_wmma_f32<<<n_nodes / 16, 256, 0, stream>>>(x, W, h);

    // Phase 2: out = bias (broadcast)
    gcn_bias_init<<<(total + 255) / 256, 256, 0, stream>>>(bias, out, total);

    // Phase 3: edge gather + atomic scatter-add (8 edges per 256-thread block)
    gcn_edge_scatter<<<(n_keep + 7) / 8, 256, 0, stream>>>(h, esrc, edst, ew,
                                                           keep, out, n_keep);
}